// IILoss_28784870818528
// MI455X (gfx1250) — compile-verified
//
#include <hip/hip_runtime.h>
#include <hip/hip_bf16.h>

typedef __attribute__((ext_vector_type(2))) float v2f;
typedef __attribute__((ext_vector_type(8))) float v8f;

#define N_INST 262144
#define C_CLS 256
#define D_EMB 128
#define B1 256                                // blocks in pass 1
#define T1 256                                // threads per block in pass 1 (8 waves)
#define ACC_FLOATS (C_CLS * D_EMB + C_CLS)    // 33024 floats: sums + counts

// Workspace layout (as float*):
//   [0,      32768) : class sums           S[c*128+d]
//   [32768,  33024) : class counts (float) CNT[c]
//   [33024,  65792) : centers mu           MU[c*128+d]
//   [65792,  66048) : per-class ||mu||^2   NRM[c]
//   byte 264192     : double ssq, double wsum, uint32 minbits
#define OFF_CNT  (C_CLS * D_EMB)
#define OFF_MU   33024
#define OFF_NRM  65792
#define OFF_DBL  66048

__global__ void k_init(float* __restrict__ ws) {
    int idx = blockIdx.x * blockDim.x + threadIdx.x;
    if (idx < ACC_FLOATS) ws[idx] = 0.0f;
    if (idx == 0) {
        double* dd = (double*)(ws + OFF_DBL);
        dd[0] = 0.0;                       // ssq
        dd[1] = 0.0;                       // wsum
        *(unsigned*)(dd + 2) = 0x7F800000u; // minbits = +inf
    }
}

// Pass 1: one sweep over embeddings. Per-class sums + counts accumulated in
// LDS (ds_add_f32), sum of squares reduced per-wave, f64 atomic at the end.
__global__ void k_accum(const float* __restrict__ emb,
                        const int* __restrict__ tgt,
                        float* __restrict__ ws) {
    extern __shared__ float lds[];
    const int tid = threadIdx.x;
    for (int k = tid; k < ACC_FLOATS; k += T1) lds[k] = 0.0f;
    __syncthreads();

    const int wave = tid >> 5;
    const int lane = tid & 31;
    const int perBlock = N_INST / B1;          // 1024 instances per block
    const int start = blockIdx.x * perBlock;
    float ssq = 0.0f;

    for (int i = start + wave; i < start + perBlock; i += (T1 / 32)) {
        const int t = tgt[i];
        const float4 x = *(const float4*)(emb + (size_t)i * D_EMB + lane * 4);
        float* dst = lds + t * D_EMB + lane * 4;
        atomicAdd(dst + 0, x.x);
        atomicAdd(dst + 1, x.y);
        atomicAdd(dst + 2, x.z);
        atomicAdd(dst + 3, x.w);
        ssq += x.x * x.x + x.y * x.y + x.z * x.z + x.w * x.w;
        if (lane == 0) atomicAdd(lds + OFF_CNT + t, 1.0f);
    }
    __syncthreads();

    // Flush LDS accumulators to global (skip zeros to cut atomic traffic)
    for (int k = tid; k < ACC_FLOATS; k += T1) {
        float v = lds[k];
        if (v != 0.0f) atomicAdd(ws + k, v);
    }

    // Reduce sum-of-squares: wave shuffle, then cross-wave via LDS
    for (int off = 16; off >= 1; off >>= 1) ssq += __shfl_down(ssq, off, 32);
    __syncthreads();
    if (lane == 0) lds[wave] = ssq;
    __syncthreads();
    if (tid == 0) {
        float tot = 0.0f;
        for (int w = 0; w < T1 / 32; ++w) tot += lds[w];
        double* dd = (double*)(ws + OFF_DBL);
        atomicAdd(&dd[0], (double)tot);
    }
}

// Pass 2: mu = S / max(cnt,1), ||mu||^2 per class, wsum += n*||mu||^2 (f64)
__global__ void k_centers(float* __restrict__ ws) {
    const int c = blockIdx.x;      // 256 blocks
    const int d = threadIdx.x;     // 128 threads = 4 waves
    __shared__ float red[4];
    const float n = ws[OFF_CNT + c];
    const float inv = 1.0f / fmaxf(n, 1.0f);
    float m = ws[c * D_EMB + d] * inv;
    ws[OFF_MU + c * D_EMB + d] = m;
    float sq = m * m;
    for (int off = 16; off >= 1; off >>= 1) sq += __shfl_down(sq, off, 32);
    if ((d & 31) == 0) red[d >> 5] = sq;
    __syncthreads();
    if (d == 0) {
        float n2 = red[0] + red[1] + red[2] + red[3];
        ws[OFF_NRM + c] = n2;
        double* dd = (double*)(ws + OFF_DBL);
        atomicAdd(&dd[1], (double)(n * n2));
    }
}

// Pass 3: Gram = mu @ mu^T via V_WMMA_F32_16X16X4_F32; one 16x16 tile per
// wave, K=128 in 32 steps. dist = nrm[a]+nrm[b]-2*G; diag -> 1e24; min-reduce.
__global__ void k_gram_min(float* __restrict__ ws) {
    const float* MU  = ws + OFF_MU;
    const float* NRM = ws + OFF_NRM;
    const int tid  = threadIdx.x;
    const int lane = tid & 31;
    const int g    = blockIdx.x * (blockDim.x >> 5) + (tid >> 5); // tile 0..255
    const int tm   = g >> 4;
    const int tn   = g & 15;
    const int lhi  = lane >> 4;     // 0 or 1
    const int llo  = lane & 15;

    // 32-bit A 16x4 layout: lane l, VGPR v holds A[M = l%16][K = (l/16)*2 + v]
    const float* arow = MU + (size_t)(tm * 16 + llo) * D_EMB + lhi * 2;
    const float* brow = MU + (size_t)(tn * 16 + llo) * D_EMB + lhi * 2;

    v8f acc = {};
#pragma unroll
    for (int k0 = 0; k0 < D_EMB; k0 += 4) {
        v2f a = *(const v2f*)(arow + k0);
        v2f b = *(const v2f*)(brow + k0);
        acc = __builtin_amdgcn_wmma_f32_16x16x4_f32(
            /*neg_a=*/false, a, /*neg_b=*/false, b,
            /*c_mod=*/(short)0, acc, /*reuse_a=*/false, /*reuse_b=*/false);
    }

    // C/D layout: VGPR r, lanes 0-15 -> M=r, lanes 16-31 -> M=r+8, N=lane%16
    float mn = 3.0e38f;
#pragma unroll
    for (int r = 0; r < 8; ++r) {
        int row = tm * 16 + r + lhi * 8;
        int col = tn * 16 + llo;
        float dist = NRM[row] + NRM[col] - 2.0f * acc[r];
        dist = fmaxf(dist, 0.0f);
        if (row == col) dist = 1.0e24f;
        mn = fminf(mn, dist);
    }
    for (int off = 16; off >= 1; off >>= 1)
        mn = fminf(mn, __shfl_xor(mn, off, 32));
    if (lane == 0) {
        unsigned* mb = (unsigned*)((double*)(ws + OFF_DBL) + 2);
        atomicMin(mb, __float_as_uint(mn)); // dists >= 0: bit order == fp order
    }
}

__global__ void k_final(const float* __restrict__ ws, float* __restrict__ out) {
    const double* dd = (const double*)(ws + OFF_DBL);
    const unsigned mb = *(const unsigned*)(dd + 2);
    out[0] = (float)((dd[0] - dd[1]) / (double)N_INST); // intra_spread
    out[1] = -__uint_as_float(mb);                      // inter_separation
}

extern "C" void kernel_launch(void* const* d_in, const int* in_sizes, int n_in,
                              void* d_out, int out_size, void* d_ws, size_t ws_size,
                              hipStream_t stream) {
    const float* emb = (const float*)d_in[0];
    const int*   tgt = (const int*)d_in[1];
    float* ws  = (float*)d_ws;
    float* out = (float*)d_out;

    k_init<<<(ACC_FLOATS + 255) / 256, 256, 0, stream>>>(ws);
    k_accum<<<B1, T1, ACC_FLOATS * sizeof(float), stream>>>(emb, tgt, ws);
    k_centers<<<C_CLS, D_EMB, 0, stream>>>(ws);
    k_gram_min<<<32, 256, 0, stream>>>(ws);
    k_final<<<1, 1, 0, stream>>>(ws, out);
}